// Graph_OurConvNet_88192858456480
// MI455X (gfx1250) — compile-verified
//
#include <hip/hip_runtime.h>
#include <hip/hip_bf16.h>
#include <math.h>

// ---------------------------------------------------------------------------
// Graph gated-GCN forward for MI455X (gfx1250, wave32, WMMA).
// fp32 throughout: GEMM FLOPs are trivial (~18 GFLOP) vs 23.3 TB/s HBM /
// 192 MB L2, so we use V_WMMA_F32_16X16X4_F32 for exact fp32 matrix math.
// Edge phase (the bandwidth-dominant part) uses b128 gathers + f32 atomics
// that resolve in L2 (all node tensors ~77 MB << 192 MB L2).
// ---------------------------------------------------------------------------

typedef float v2f __attribute__((ext_vector_type(2)));
typedef float v8f __attribute__((ext_vector_type(8)));

#define N_NODES_C 50000
#define N_EDGES_C 800000
#define HDIM      64
#define LC_C      5
#define NB        (N_NODES_C * HDIM)   // floats per node-feature buffer

// ---------------------------------------------------------------------------
// Embedding gather (vectorized): x[n][q*4..q*4+3] = emb[signal[n]][...]
// ---------------------------------------------------------------------------
__global__ __launch_bounds__(256) void embed_kernel(const int* __restrict__ signal,
                                                    const float4* __restrict__ emb4,
                                                    float4* __restrict__ x4, int n) {
  int t = blockIdx.x * 256 + threadIdx.x;
  if (t >= n * 16) return;
  int row = t >> 4;
  int q   = t & 15;
  x4[t] = emb4[signal[row] * 16 + q];
}

// ---------------------------------------------------------------------------
// Multi-matrix GEMM: out[m] = X @ W[m]^T (+ bias[m]), W[m] is [64][64] row-major.
// One wave computes a 16-row tile for all matrices / column tiles via
// V_WMMA_F32_16X16X4_F32, K-loop of 16. Weights staged in LDS (stride 66:
// conflict-free banks, keeps 8-byte alignment for ds_load_b64 fragments).
// ---------------------------------------------------------------------------
struct GemmArgs {
  const float* X;
  const float* W[5];
  const float* bias[5];   // nullptr => no bias
  float*       out[5];
  int          nmat;
  int          n;         // number of rows (nodes)
};

__global__ __launch_bounds__(256) void gemm_multi(GemmArgs args) {
  __shared__ float lds_w[5 * 64 * 66];   // 84,480 B of the 320 KB WGP LDS

  const int tid = threadIdx.x;
  // Cooperative, fully uniform weight load (nmat*4096 is a multiple of 256).
  const int total = args.nmat * 4096;
  for (int idx = tid; idx < total; idx += 256) {
    int mat = idx >> 12;
    int rem = idx & 4095;
    int h   = rem >> 6;
    int d   = rem & 63;
    lds_w[(mat * 64 + h) * 66 + d] = args.W[mat][rem];
  }
  __syncthreads();

  const int wave = tid >> 5;
  const int lane = tid & 31;
  const int m    = lane & 15;   // row-in-tile (A) / col-in-tile (B,C)
  const int hl   = lane >> 4;   // half-wave select
  const int row0 = blockIdx.x * 128 + wave * 16;

  // Preload A fragments for this wave's 16-row tile (clamped; stores guarded).
  int rload = row0 + m;
  rload = rload < args.n ? rload : (args.n - 1);
  v2f a[16];
  const float* xr = args.X + rload * HDIM + 2 * hl;
#pragma unroll
  for (int kk = 0; kk < 16; ++kk)
    a[kk] = *(const v2f*)(xr + kk * 4);

  for (int mat = 0; mat < args.nmat; ++mat) {
    const float* bptr = args.bias[mat];
    float*       optr = args.out[mat];
#pragma unroll
    for (int ct = 0; ct < 4; ++ct) {
      const int col = ct * 16 + m;
      const float* bl = &lds_w[(mat * 64 + col) * 66 + 2 * hl];
      v8f acc = {0.f, 0.f, 0.f, 0.f, 0.f, 0.f, 0.f, 0.f};
#pragma unroll
      for (int kk = 0; kk < 16; ++kk) {
        v2f bf = *(const v2f*)(bl + kk * 4);
        acc = __builtin_amdgcn_wmma_f32_16x16x4_f32(
            /*neg_a=*/false, a[kk], /*neg_b=*/false, bf,
            /*c_mod=*/(short)0, acc, /*reuse_a=*/false, /*reuse_b=*/false);
      }
      float bias = bptr ? bptr[col] : 0.0f;
#pragma unroll
      for (int v = 0; v < 8; ++v) {
        int r = row0 + hl * 8 + v;
        if (r < args.n) optr[r * HDIM + col] = acc[v] + bias;
      }
    }
  }
}

// ---------------------------------------------------------------------------
// Edge phase: agg[ee] += sigmoid(viX[ee] + vjX[es]) * ujX[es]
// (bv folded into viX, bu + x@Ui^T already in agg).
// 16 threads/edge, float4 (global_load_b128) gathers, f32 atomics into L2.
// ---------------------------------------------------------------------------
__global__ __launch_bounds__(256) void edge_kernel(const int* __restrict__ es,
                                                   const int* __restrict__ ee,
                                                   const float4* __restrict__ viX,
                                                   const float4* __restrict__ vjX,
                                                   const float4* __restrict__ ujX,
                                                   float* __restrict__ agg, int E) {
  int t = blockIdx.x * 256 + threadIdx.x;
  int e = t >> 4;
  if (e >= E) return;
  int q = t & 15;                 // which float4 of the 16 per 64-float row
  int a = ee[e];
  int b = es[e];
  float4 vi = viX[a * 16 + q];
  float4 vj = vjX[b * 16 + q];
  float4 uj = ujX[b * 16 + q];
  float* dst = agg + a * HDIM + q * 4;
  float g0 = 1.0f / (1.0f + __expf(-(vi.x + vj.x)));
  float g1 = 1.0f / (1.0f + __expf(-(vi.y + vj.y)));
  float g2 = 1.0f / (1.0f + __expf(-(vi.z + vj.z)));
  float g3 = 1.0f / (1.0f + __expf(-(vi.w + vj.w)));
  atomicAdd(dst + 0, g0 * uj.x);
  atomicAdd(dst + 1, g1 * uj.y);
  atomicAdd(dst + 2, g2 * uj.z);
  atomicAdd(dst + 3, g3 * uj.w);
}

// ---------------------------------------------------------------------------
// BatchNorm statistics: stats[0..63] = col sums, stats[64..127] = col sumsq.
// ---------------------------------------------------------------------------
__global__ __launch_bounds__(128) void zero_stats(float* stats) {
  stats[threadIdx.x] = 0.0f;
}

__global__ __launch_bounds__(256) void bn_stats(const float* __restrict__ x,
                                                float* __restrict__ stats, int n) {
  __shared__ float red[512];
  int col  = threadIdx.x & 63;
  int rgrp = threadIdx.x >> 6;  // 0..3
  float s = 0.f, ss = 0.f;
  for (int r = blockIdx.x * 4 + rgrp; r < n; r += gridDim.x * 4) {
    float v = x[r * HDIM + col];
    s  += v;
    ss += v * v;
  }
  red[threadIdx.x]       = s;
  red[256 + threadIdx.x] = ss;
  __syncthreads();
  if (threadIdx.x < 64) {
    s  = red[col] + red[64 + col] + red[128 + col] + red[192 + col];
    ss = red[256 + col] + red[320 + col] + red[384 + col] + red[448 + col];
    atomicAdd(&stats[col], s);
    atomicAdd(&stats[64 + col], ss);
  }
}

// Fused BN apply (+ optional residual, + optional ReLU).
__global__ __launch_bounds__(256) void bn_apply(const float* __restrict__ x,
                                                const float* __restrict__ stats,
                                                const float* __restrict__ g,
                                                const float* __restrict__ b,
                                                const float* __restrict__ res,
                                                float* __restrict__ y,
                                                int n, int relu) {
  int t = blockIdx.x * 256 + threadIdx.x;
  int row = t >> 6;
  if (row >= n) return;
  int col = t & 63;
  float inv_n = 1.0f / (float)n;
  float mean  = stats[col] * inv_n;
  float var   = stats[64 + col] * inv_n - mean * mean;
  float sc    = g[col] * rsqrtf(var + 1e-5f);
  float v = (x[row * HDIM + col] - mean) * sc + b[col];
  if (res)  v += res[row * HDIM + col];
  if (relu) v = fmaxf(v, 0.0f);
  y[row * HDIM + col] = v;
}

// ---------------------------------------------------------------------------
// Final FC: out[n][c] = x[n] . fc_w[c] + fc_b[c]  (8 clusters)
// ---------------------------------------------------------------------------
__global__ __launch_bounds__(256) void fc_kernel(const float* __restrict__ x,
                                                 const float* __restrict__ w,
                                                 const float* __restrict__ bias,
                                                 float* __restrict__ out, int n) {
  int t = blockIdx.x * 256 + threadIdx.x;
  int row = t >> 3;
  if (row >= n) return;
  int c = t & 7;
  const float* xr = x + row * HDIM;
  const float* wr = w + c * HDIM;
  float acc = bias[c];
#pragma unroll
  for (int d = 0; d < HDIM; ++d) acc = fmaf(xr[d], wr[d], acc);
  out[row * 8 + c] = acc;
}

// ---------------------------------------------------------------------------
extern "C" void kernel_launch(void* const* d_in, const int* in_sizes, int n_in,
                              void* d_out, int out_size, void* d_ws, size_t ws_size,
                              hipStream_t stream) {
  const int*   signal = (const int*)d_in[0];
  const int*   es     = (const int*)d_in[1];
  const int*   ee     = (const int*)d_in[2];
  const float* emb    = (const float*)d_in[3];
  const float* Ui1    = (const float*)d_in[4];
  const float* Uj1    = (const float*)d_in[5];
  const float* Vi1    = (const float*)d_in[6];
  const float* Vj1    = (const float*)d_in[7];
  const float* Ui2    = (const float*)d_in[8];
  const float* Uj2    = (const float*)d_in[9];
  const float* Vi2    = (const float*)d_in[10];
  const float* Vj2    = (const float*)d_in[11];
  const float* Rw     = (const float*)d_in[12];
  const float* bu1    = (const float*)d_in[13];
  const float* bv1    = (const float*)d_in[14];
  const float* bu2    = (const float*)d_in[15];
  const float* bv2    = (const float*)d_in[16];
  const float* g1     = (const float*)d_in[17];
  const float* b1     = (const float*)d_in[18];
  const float* g2     = (const float*)d_in[19];
  const float* b2     = (const float*)d_in[20];
  const float* fcw    = (const float*)d_in[21];
  const float* fcb    = (const float*)d_in[22];

  float* B0    = (float*)d_ws;   // x / vi2X scratch
  float* B1    = B0 + NB;        // uiX (+bu) -> agg -> BN input
  float* B2    = B1 + NB;        // ujX
  float* B3    = B2 + NB;        // viX (+bv) -> x2 (hidden after BN1)
  float* B4    = B3 + NB;        // vjX
  float* B5    = B4 + NB;        // residual (xin @ Rw^T)
  float* stats = B5 + NB;        // 128 floats

  const int node_blocks  = (N_NODES_C * HDIM + 255) / 256;
  const int node4_blocks = (N_NODES_C * 16 + 255) / 256;
  const int edge_blocks  = (N_EDGES_C * 16 + 255) / 256;
  const dim3 gemm_grid((N_NODES_C + 127) / 128);

  embed_kernel<<<node4_blocks, 256, 0, stream>>>(signal, (const float4*)emb,
                                                 (float4*)B0, N_NODES_C);

  for (int l = 0; l < LC_C; ++l) {
    // ---- gated block 1 (+ residual projection Rw fused into this GEMM) ----
    GemmArgs ga;
    ga.X = B0;
    ga.W[0] = Ui1 + l * 4096; ga.W[1] = Uj1 + l * 4096;
    ga.W[2] = Vi1 + l * 4096; ga.W[3] = Vj1 + l * 4096;
    ga.W[4] = Rw  + l * 4096;
    ga.bias[0] = bu1 + l * 64; ga.bias[1] = nullptr;
    ga.bias[2] = bv1 + l * 64; ga.bias[3] = nullptr; ga.bias[4] = nullptr;
    ga.out[0] = B1; ga.out[1] = B2; ga.out[2] = B3; ga.out[3] = B4; ga.out[4] = B5;
    ga.nmat = 5; ga.n = N_NODES_C;
    gemm_multi<<<gemm_grid, 256, 0, stream>>>(ga);

    edge_kernel<<<edge_blocks, 256, 0, stream>>>(es, ee, (const float4*)B3,
                                                 (const float4*)B4,
                                                 (const float4*)B2, B1, N_EDGES_C);

    zero_stats<<<1, 128, 0, stream>>>(stats);
    bn_stats<<<256, 256, 0, stream>>>(B1, stats, N_NODES_C);
    bn_apply<<<node_blocks, 256, 0, stream>>>(B1, stats, g1 + l * 64, b1 + l * 64,
                                              nullptr, B3, N_NODES_C, 1);

    // ---- gated block 2 ----
    GemmArgs gb;
    gb.X = B3;
    gb.W[0] = Ui2 + l * 4096; gb.W[1] = Uj2 + l * 4096;
    gb.W[2] = Vi2 + l * 4096; gb.W[3] = Vj2 + l * 4096; gb.W[4] = nullptr;
    gb.bias[0] = bu2 + l * 64; gb.bias[1] = nullptr;
    gb.bias[2] = bv2 + l * 64; gb.bias[3] = nullptr; gb.bias[4] = nullptr;
    gb.out[0] = B1; gb.out[1] = B2; gb.out[2] = B0; gb.out[3] = B4; gb.out[4] = nullptr;
    gb.nmat = 4; gb.n = N_NODES_C;
    gemm_multi<<<gemm_grid, 256, 0, stream>>>(gb);

    edge_kernel<<<edge_blocks, 256, 0, stream>>>(es, ee, (const float4*)B0,
                                                 (const float4*)B4,
                                                 (const float4*)B2, B1, N_EDGES_C);

    zero_stats<<<1, 128, 0, stream>>>(stats);
    bn_stats<<<256, 256, 0, stream>>>(B1, stats, N_NODES_C);
    // BN2 + residual + ReLU -> next cell input in B0
    bn_apply<<<node_blocks, 256, 0, stream>>>(B1, stats, g2 + l * 64, b2 + l * 64,
                                              B5, B0, N_NODES_C, 1);
  }

  fc_kernel<<<(N_NODES_C * 8 + 255) / 256, 256, 0, stream>>>(B0, fcw, fcb,
                                                             (float*)d_out, N_NODES_C);
}